// ODEFunc_65403761983979
// MI455X (gfx1250) — compile-verified
//
#include <hip/hip_runtime.h>
#include <math.h>
#include <stdint.h>

// ---------------------------------------------------------------------------
// MI455X reasoning: six SpMM passes over random sparse operators; 18.8M
// nonzeros x (256B gather + 256B f32-atomic scatter) ~= 12 GB/call
// => ~0.5 ms floor at 23.3 TB/s. Compute ~2.4 GFLOP => pure bandwidth/latency
// problem; WMMA does not apply (scalar weight * vector row).
// CDNA5 paths used:
//   - global_load_async_to_lds_b128 (ASYNCcnt) double-buffered staging of the
//     contiguous (row,col,val) metadata stream, one instruction per 32-nnz tile
//   - s_wait_asynccnt pipelining (async loads complete in-order)
//   - ds_load_b32 broadcast reads of staged metadata (direct __shared__
//     indexing so clang keeps addrspace(3) and emits DS ops, not flat)
//   - global_prefetch_b8: after a tile's indices land in LDS, all 32 gather
//     rows are prefetched at once (one row per lane, both 128B halves)
//   - global_atomic_add_f32 via unsafeAtomicAdd for the scatter
// ---------------------------------------------------------------------------

#define HIDV   64      // hidden dim: 16 lanes/nonzero * float4/lane
#define TILE   32      // nonzeros staged per wave per async op
#define BLK    256
#define NWAVES (BLK / 32)

__global__ void zero_f4_kernel(float4* __restrict__ p, long n4) {
  long i = (long)blockIdx.x * blockDim.x + threadIdx.x;
  long stride = (long)gridDim.x * blockDim.x;
  float4 z; z.x = 0.f; z.y = 0.f; z.z = 0.f; z.w = 0.f;
  for (; i < n4; i += stride) p[i] = z;
}

// Low 32 bits of a flat shared-aperture address == LDS byte offset.
__device__ __forceinline__ uint32_t lds_off_u32(const void* p) {
  return (uint32_t)(uintptr_t)p;
}

// Stage one 32-nonzero tile of metadata into this wave's LDS slot with a
// single per-lane async b128 copy: lanes 0-7 -> oidx, 8-15 -> gidx,
// 16-23 -> vals (each lane moves 16 bytes). Tracked by ASYNCcnt.
__device__ __forceinline__ void async_stage_tile(const int* __restrict__ oidx,
                                                 const int* __restrict__ gidx,
                                                 const float* __restrict__ vals,
                                                 long kbase, int* buf, int lane) {
  // Ensure prior DS reads of this buffer drained before the async engine
  // overwrites it (cheap; DScnt is normally already 0 here).
  asm volatile("s_wait_dscnt 0x0" ::: "memory");
  if (lane < 24) {
    const int section = lane >> 3;        // 0:oidx 1:gidx 2:vals
    const int e       = (lane & 7) << 2;  // dword index within tile: 0,4,...,28
    const void* src = (section == 0) ? (const void*)(oidx + kbase + e)
                    : (section == 1) ? (const void*)(gidx + kbase + e)
                                     : (const void*)(vals + kbase + e);
    uint64_t gaddr = (uint64_t)(uintptr_t)src;
    uint32_t dst   = lds_off_u32(buf + section * TILE + e);  // escapes smem to the asm
    asm volatile("global_load_async_to_lds_b128 %0, %1, off"
                 :: "v"(dst), "v"(gaddr) : "memory");
  }
}

__device__ __forceinline__ void wait_async_le1() {
  asm volatile("s_wait_asynccnt 0x1" ::: "memory");
}
__device__ __forceinline__ void wait_async_0() {
  asm volatile("s_wait_asynccnt 0x0" ::: "memory");
}

// One nonzero: out[o*64 + sub*4 .. +3] += beta * v * scales * x[g*64 + ...]
// GS: gather-index scale (0 none, 1 *gsc[g], 2 /gsc[g]); OS: output-index
// scale (0 none, 2 /osc[o]).
template <int GS, int OS>
__device__ __forceinline__ void scatter_one(int o, int g, float v, float beta,
                                            const float* __restrict__ x,
                                            const float* __restrict__ gsc,
                                            const float* __restrict__ osc,
                                            float* __restrict__ out, int sub) {
  float s = beta * v;
  if (GS == 1) s *= gsc[g];
  if (GS == 2) s /= gsc[g];
  if (OS == 2) s /= osc[o];
  const float4 xv = *(const float4*)(x + (long)g * HIDV + sub * 4);
  float* op = out + (long)o * HIDV + sub * 4;
  unsafeAtomicAdd(op + 0, s * xv.x);
  unsafeAtomicAdd(op + 1, s * xv.y);
  unsafeAtomicAdd(op + 2, s * xv.z);
  unsafeAtomicAdd(op + 3, s * xv.w);
}

template <int GS, int OS, bool UA>
__global__ void __launch_bounds__(BLK)
spmm_scatter_kernel(const int*   __restrict__ oidx,
                    const int*   __restrict__ gidx,
                    const float* __restrict__ vals,
                    const float* __restrict__ x,
                    const float* __restrict__ gsc,
                    const float* __restrict__ osc,
                    const float* __restrict__ alpha_raw,
                    float*       __restrict__ out,
                    int nnz) {
  // Per-wave double-buffered metadata slots: [wave][buf][o(32) g(32) v(32)]
  __shared__ __align__(16) int smem[NWAVES][2][3 * TILE];

  const int tid  = threadIdx.x;
  const int wave = tid >> 5;
  const int lane = tid & 31;
  const int grp  = lane >> 4;   // two 16-lane groups per wave
  const int sub  = lane & 15;   // float4 slot within the 64-float row

  float beta = 1.0f;
  if (UA) beta = -1.0f / (1.0f + expf(-alpha_raw[0]));

  const long ntiles  = (long)nnz / TILE;  // full tiles only
  const long tstride = (long)gridDim.x * NWAVES;
  long t = (long)blockIdx.x * NWAVES + wave;

  int buf = 0;
  if (t < ntiles) async_stage_tile(oidx, gidx, vals, t * TILE, &smem[wave][buf][0], lane);

  for (; t < ntiles; t += tstride) {
    long tn = t + tstride;
    if (tn < ntiles) {
      async_stage_tile(oidx, gidx, vals, tn * TILE, &smem[wave][buf ^ 1][0], lane);
      wait_async_le1();   // in-order completion => tile t has landed
    } else {
      wait_async_0();
    }

    // Deep prefetch: each lane prefetches one of the tile's 32 gather rows
    // (both 128B halves of the 256B row). Direct __shared__ indexing keeps
    // addrspace(3) => ds_load_b32.
    {
      const int pg = smem[wave][buf][TILE + lane];
      const float* pr = x + (long)pg * HIDV;
      __builtin_prefetch(pr, 0, 3);
      __builtin_prefetch(pr + 32, 0, 3);
    }

#pragma unroll 4
    for (int j = 0; j < 16; ++j) {
      const int idx = grp * 16 + j;
      const int   o = smem[wave][buf][idx];
      const int   g = smem[wave][buf][TILE + idx];
      const float v = ((const float*)&smem[wave][buf][2 * TILE])[idx];
      scatter_one<GS, OS>(o, g, v, beta, x, gsc, osc, out, sub);
    }
    buf ^= 1;
  }

  // Tail (nnz % 32): handled scalar-style by block 0 / wave 0.
  const long kt = ntiles * TILE;
  if (blockIdx.x == 0 && wave == 0) {
    for (long k = kt + grp; k < nnz; k += 2)
      scatter_one<GS, OS>(oidx[k], gidx[k], vals[k], beta, x, gsc, osc, out, sub);
  }
}

static inline int grid_for_zero(long threads, int block) {
  long g = (threads + block - 1) / block;
  if (g > 65536) g = 65536;
  if (g < 1) g = 1;
  return (int)g;
}

static inline int grid_for_scatter(int nnz) {
  long ntiles = (long)nnz / TILE;
  long blocks = (ntiles + (long)NWAVES * 4 - 1) / ((long)NWAVES * 4);  // ~4 tiles/wave
  if (blocks < 1) blocks = 1;
  if (blocks > 65535) blocks = 65535;
  return (int)blocks;
}

extern "C" void kernel_launch(void* const* d_in, const int* in_sizes, int n_in,
                              void* d_out, int out_size, void* d_ws, size_t ws_size,
                              hipStream_t stream) {
  const float* q         = (const float*)d_in[0];
  const float* p         = (const float*)d_in[1];
  const float* A0        = (const float*)d_in[2];
  const float* A1        = (const float*)d_in[3];
  const float* A2        = (const float*)d_in[4];
  const float* alpha_raw = (const float*)d_in[5];
  const int*   d0r       = (const int*)d_in[6];
  const int*   d0c       = (const int*)d_in[7];
  const float* d0v       = (const float*)d_in[8];
  const int*   d1r       = (const int*)d_in[9];
  const int*   d1c       = (const int*)d_in[10];
  const float* d1v       = (const float*)d_in[11];

  const long n_nodes = in_sizes[2];
  const long n_edges = in_sizes[3];
  const long n_tri   = in_sizes[4];
  const int  nnz0    = in_sizes[6];
  const int  nnz1    = in_sizes[9];
  const int  hid     = (int)(in_sizes[0] / n_nodes);  // 64

  float* qPart = (float*)d_out;           // n_nodes * hid
  float* pPart = qPart + n_nodes * hid;   // n_edges * hid
  float* tmp   = (float*)d_ws;            // reused: E*hid -> T*hid -> N*hid

  // Zero outputs (harness poisons d_out).
  {
    long n4 = (n_nodes + n_edges) * hid / 4;
    zero_f4_kernel<<<grid_for_zero(n4, BLK), BLK, 0, stream>>>((float4*)d_out, n4);
  }

  // ---- qPart = -alpha * D0^T( a1 .* D0(q ./ a0) ) ./ a0 -------------------
  {
    long n4 = n_edges * hid / 4;
    zero_f4_kernel<<<grid_for_zero(n4, BLK), BLK, 0, stream>>>((float4*)tmp, n4);
  }
  // tmp_e[d0r[k]] += v * q[d0c[k]] / a0[d0c[k]]
  spmm_scatter_kernel<2, 0, false><<<grid_for_scatter(nnz0), BLK, 0, stream>>>(
      d0r, d0c, d0v, q, A0, nullptr, nullptr, tmp, nnz0);
  // qPart[d0c[k]] += -alpha * v * a1[d0r[k]] * tmp_e[d0r[k]] / a0[d0c[k]]
  spmm_scatter_kernel<1, 2, true><<<grid_for_scatter(nnz0), BLK, 0, stream>>>(
      d0c, d0r, d0v, tmp, A1, A0, alpha_raw, qPart, nnz0);

  // ---- pPart += -alpha * D1^T( a2 .* D1(p ./ a1) ) ./ a1 ------------------
  {
    long n4 = n_tri * hid / 4;
    zero_f4_kernel<<<grid_for_zero(n4, BLK), BLK, 0, stream>>>((float4*)tmp, n4);
  }
  // tmp_t[d1r[k]] += v * p[d1c[k]] / a1[d1c[k]]
  spmm_scatter_kernel<2, 0, false><<<grid_for_scatter(nnz1), BLK, 0, stream>>>(
      d1r, d1c, d1v, p, A1, nullptr, nullptr, tmp, nnz1);
  // pPart[d1c[k]] += -alpha * v * a2[d1r[k]] * tmp_t[d1r[k]] / a1[d1c[k]]
  spmm_scatter_kernel<1, 2, true><<<grid_for_scatter(nnz1), BLK, 0, stream>>>(
      d1c, d1r, d1v, tmp, A2, A1, alpha_raw, pPart, nnz1);

  // ---- pPart += -alpha * D0( (D0^T p) ./ a0 ) -----------------------------
  {
    long n4 = n_nodes * hid / 4;
    zero_f4_kernel<<<grid_for_zero(n4, BLK), BLK, 0, stream>>>((float4*)tmp, n4);
  }
  // tmp_n[d0c[k]] += v * p[d0r[k]]
  spmm_scatter_kernel<0, 0, false><<<grid_for_scatter(nnz0), BLK, 0, stream>>>(
      d0c, d0r, d0v, p, nullptr, nullptr, nullptr, tmp, nnz0);
  // pPart[d0r[k]] += -alpha * v * tmp_n[d0c[k]] / a0[d0c[k]]
  spmm_scatter_kernel<2, 0, true><<<grid_for_scatter(nnz0), BLK, 0, stream>>>(
      d0r, d0c, d0v, tmp, A0, nullptr, alpha_raw, pPart, nnz0);
}